// Method_Pubmed_63144609185882
// MI455X (gfx1250) — compile-verified
//
#include <hip/hip_runtime.h>

typedef float v2f __attribute__((ext_vector_type(2)));
typedef float v8f __attribute__((ext_vector_type(8)));

constexpr int IN_DIM  = 500;
constexpr int HID     = 8;
constexpr int OUTD    = 3;
constexpr int LSTRIDE = 514;   // LDS row stride (floats): spreads 8 columns across banks

// ---------------------------------------------------------------------------
// Init: agg1[i][j] = b1[j], out[i][j] = b2[j]  (bias folded into segment-sum)
// ---------------------------------------------------------------------------
__global__ void gcn_init(float* __restrict__ agg1, float* __restrict__ out,
                         const float* __restrict__ b1, const float* __restrict__ b2,
                         int n8, int n3) {
    int i = blockIdx.x * blockDim.x + threadIdx.x;
    if (i < n8) agg1[i] = b1[i & (HID - 1)];
    if (i < n3) out[i]  = b2[i % OUTD];
}

// ---------------------------------------------------------------------------
// Layer-1 dense GEMM: sup1 = x @ W1 via V_WMMA_F32_16X16X4_F32.
// One wave per 16-row tile of x; K=500 consumed 4 at a time (125 WMMAs).
// A frag (16x4 f32): lane L(0-15) holds K=k0,k0+1; lane L+16 holds K=k0+2,k0+3.
// B frag (4x16 f32): mirrored layout; W1 staged transposed in LDS, column 8
// zero-filled so N-columns 8..15 contribute zero without predication.
// ---------------------------------------------------------------------------
__global__ __launch_bounds__(256) void gcn_gemm1_wmma(
        const float* __restrict__ x, const float* __restrict__ W1,
        float* __restrict__ sup1, int nNodes) {
    __shared__ float wt[9 * LSTRIDE];

    // Stage W1 transposed: wt[c][k] = W1[k][c] for c<8, wt[8][k] = 0.
    for (int i = threadIdx.x; i < 9 * IN_DIM; i += 256) {
        int c = i / IN_DIM;
        int k = i - c * IN_DIM;
        wt[c * LSTRIDE + k] = (c < HID) ? W1[k * HID + c] : 0.0f;
    }
    __syncthreads();

    const int lane = threadIdx.x & 31;
    const int wave = threadIdx.x >> 5;          // 8 waves per block
    const int tile = blockIdx.x * 8 + wave;     // 16-row tile index
    if (tile * 16 >= nNodes) return;            // wave-uniform guard

    const int col  = lane & 15;                 // M-row (A) / N-col (B,D)
    const int koff = (lane >> 4) << 1;          // 0 for lanes 0-15, 2 for 16-31

    const float* xrow = x + (long long)(tile * 16 + col) * IN_DIM + koff;
    const float* bp   = wt + (col < HID ? col : HID) * LSTRIDE + koff;

    v8f acc = {};
    #pragma unroll 5
    for (int k0 = 0; k0 < IN_DIM; k0 += 4) {
        v2f a = *(const v2f*)(xrow + k0);       // global_load_b64, 8B aligned
        v2f b = *(const v2f*)(bp + k0);         // ds_load_b64, 8B aligned
        acc = __builtin_amdgcn_wmma_f32_16x16x4_f32(
                  false, a, false, b, (short)0, acc, false, false);
    }

    // D layout: lanes 0-15 VGPR r -> (M=r, N=lane); lanes 16-31 -> (M=8+r, N=lane-16)
    if (col < HID) {
        long long mbase = (long long)tile * 16 + ((lane >> 4) << 3);
        #pragma unroll
        for (int r = 0; r < 8; ++r)
            sup1[(mbase + r) * HID + col] = acc[r];
    }
}

// ---------------------------------------------------------------------------
// Edge scatter, layer 1: agg1[dst] += sup1[src]  (8 fp32 atomics / edge)
// sup1/agg1 are 8 MB each -> resident in 192 MB L2; atomics resolve at L2.
// ---------------------------------------------------------------------------
__global__ void gcn_scatter_h(const long long* __restrict__ src,
                              const long long* __restrict__ dst,
                              const float* __restrict__ sup,
                              float* __restrict__ agg, int nE) {
    int e = blockIdx.x * blockDim.x + threadIdx.x;
    if (e >= nE) return;
    long long s = src[e];
    long long d = dst[e];
    const float4* sp = (const float4*)(sup + s * HID);   // 32B aligned
    float4 v0 = sp[0];
    float4 v1 = sp[1];
    float* ap = agg + d * HID;
    unsafeAtomicAdd(ap + 0, v0.x);
    unsafeAtomicAdd(ap + 1, v0.y);
    unsafeAtomicAdd(ap + 2, v0.z);
    unsafeAtomicAdd(ap + 3, v0.w);
    unsafeAtomicAdd(ap + 4, v1.x);
    unsafeAtomicAdd(ap + 5, v1.y);
    unsafeAtomicAdd(ap + 6, v1.z);
    unsafeAtomicAdd(ap + 7, v1.w);
}

// ---------------------------------------------------------------------------
// Layer-2 dense: sup2 = relu(agg1) @ W2   (K=8, N=3 -> VALU, bandwidth bound)
// agg1 already contains b1 + segment_sum.
// ---------------------------------------------------------------------------
__global__ void gcn_layer2(const float* __restrict__ agg,
                           const float* __restrict__ W2,
                           float* __restrict__ sup2, int nNodes) {
    int n = blockIdx.x * blockDim.x + threadIdx.x;
    if (n >= nNodes) return;
    const float4* ap = (const float4*)(agg + (long long)n * HID);
    float4 a0 = ap[0];
    float4 a1 = ap[1];
    float h[HID] = { fmaxf(a0.x, 0.f), fmaxf(a0.y, 0.f), fmaxf(a0.z, 0.f), fmaxf(a0.w, 0.f),
                     fmaxf(a1.x, 0.f), fmaxf(a1.y, 0.f), fmaxf(a1.z, 0.f), fmaxf(a1.w, 0.f) };
    float o0 = 0.f, o1 = 0.f, o2 = 0.f;
    #pragma unroll
    for (int k = 0; k < HID; ++k) {
        o0 = fmaf(h[k], W2[k * OUTD + 0], o0);   // uniform index -> scalar loads
        o1 = fmaf(h[k], W2[k * OUTD + 1], o1);
        o2 = fmaf(h[k], W2[k * OUTD + 2], o2);
    }
    float* op = sup2 + (long long)n * OUTD;
    op[0] = o0; op[1] = o1; op[2] = o2;
}

// ---------------------------------------------------------------------------
// Edge scatter, layer 2: out[dst] += sup2[src]  (3 fp32 atomics / edge)
// ---------------------------------------------------------------------------
__global__ void gcn_scatter_o(const long long* __restrict__ src,
                              const long long* __restrict__ dst,
                              const float* __restrict__ sup,
                              float* __restrict__ out, int nE) {
    int e = blockIdx.x * blockDim.x + threadIdx.x;
    if (e >= nE) return;
    long long s = src[e];
    long long d = dst[e];
    const float* sp = sup + s * OUTD;
    float v0 = sp[0], v1 = sp[1], v2 = sp[2];
    float* op = out + d * OUTD;
    unsafeAtomicAdd(op + 0, v0);
    unsafeAtomicAdd(op + 1, v1);
    unsafeAtomicAdd(op + 2, v2);
}

// ---------------------------------------------------------------------------
extern "C" void kernel_launch(void* const* d_in, const int* in_sizes, int n_in,
                              void* d_out, int out_size, void* d_ws, size_t ws_size,
                              hipStream_t stream) {
    const float*     x  = (const float*)d_in[0];
    const long long* ei = (const long long*)d_in[1];   // int64 [2, nE]
    const float*     W1 = (const float*)d_in[2];
    const float*     b1 = (const float*)d_in[3];
    const float*     W2 = (const float*)d_in[4];
    const float*     b2 = (const float*)d_in[5];
    float* out = (float*)d_out;

    const int nNodes = in_sizes[0] / IN_DIM;           // 262144
    const int nE     = in_sizes[1] / 2;                // 8388608
    const long long* src = ei;
    const long long* dst = ei + nE;

    // Workspace: sup1 [N*8] | agg1 [N*8] | sup2 [N*3]   (~19 MB total)
    float* sup1 = (float*)d_ws;
    float* agg1 = sup1 + (size_t)nNodes * HID;
    float* sup2 = agg1 + (size_t)nNodes * HID;

    const int n8 = nNodes * HID;
    const int n3 = nNodes * OUTD;

    gcn_init<<<(n8 + 255) / 256, 256, 0, stream>>>(agg1, out, b1, b2, n8, n3);
    gcn_gemm1_wmma<<<(nNodes + 127) / 128, 256, 0, stream>>>(x, W1, sup1, nNodes);
    gcn_scatter_h<<<(nE + 255) / 256, 256, 0, stream>>>(src, dst, sup1, agg1, nE);
    gcn_layer2<<<(nNodes + 255) / 256, 256, 0, stream>>>(agg1, W2, sup2, nNodes);
    gcn_scatter_o<<<(nE + 255) / 256, 256, 0, stream>>>(src, dst, sup2, out, nE);
}